// TemporalMaskModel_87582973100392
// MI455X (gfx1250) — compile-verified
//
#include <hip/hip_runtime.h>

typedef float v2f __attribute__((ext_vector_type(2)));
typedef float v8f __attribute__((ext_vector_type(8)));

#define HWSZ 8192   // 64*128
#define Hd   64
#define Wd   128
#define EPS  1e-5f
#define KC   16     // K-chunk (input channels staged per iteration)

// ragged group offsets from COUNTS = [[3,4],[2,5],[4,3],[3,4]] (timestep-major)
__constant__ int c_seg_off[9] = {0, 3, 7, 9, 14, 18, 21, 24, 28};

// ---------------------------------------------------------------------------
// LDS-tiled implicit-GEMM conv3x3 (SAME, stride 1) via V_WMMA_F32_16X16X4_F32.
// Block (256 thr = 8 waves): one image n, one 64-outch group, one row h0
//   wave wv owns output cols [wv*16, wv*16+16) x 64 out channels.
// LDS (channel-PAIR interleaved so each fragment is one aligned ds_load_b64):
//   lds_w: [tap][ic/2][64 oc][2]   = 9*8*64*2  floats (36,864 B)
//   lds_x: [ic/2][3 rows][130][2]  = 8*3*130*2 floats (24,960 B), zero-padded
// Frag layouts (ISA): A row = lane&15, K = (lane>>4)*2 (+1);
//                     B col = lane&15, K likewise; D row r+8*(lane>>4), col lane&15
// ---------------------------------------------------------------------------
__global__ __launch_bounds__(256) void conv3x3_wmma(const float* __restrict__ x,
                                                    const float* __restrict__ w,
                                                    const float* __restrict__ bias,
                                                    float* __restrict__ out,
                                                    int Nimg, int Cin, int O) {
  const int ngrps = O >> 6;
  const int h0    = blockIdx.x % Hd;
  const int ngrp  = (blockIdx.x / Hd) % ngrps;
  const int n     = blockIdx.x / (Hd * ngrps);

  __shared__ float lds_w[9 * (KC / 2) * 64 * 2];   // 9216 floats
  __shared__ float lds_x[(KC / 2) * 3 * 130 * 2];  // 6240 floats

  const int tid     = threadIdx.x;
  const int lane    = tid & 31;
  const int wv      = tid >> 5;
  const int colc    = lane & 15;
  const int half    = lane >> 4;       // 0/1: selects K sub-pair of the fragment
  const int colbase = wv * 16;

  const float* xn    = x + (size_t)n * Cin * HWSZ;
  const size_t wbase = (size_t)(ngrp * 64) * Cin * 9;

  v8f acc0 = 0, acc1 = 0, acc2 = 0, acc3 = 0;

  for (int kc0 = 0; kc0 < Cin; kc0 += KC) {
    __syncthreads();   // protect LDS reuse across chunks

    // ---- stage weights: 64 oc x KC ic x 9 taps (source-contiguous reads) ----
    for (int e = tid; e < 64 * KC * 9; e += 256) {
      const int oc  = e / (KC * 9);
      const int rem = e - oc * (KC * 9);
      const int ic  = rem / 9;
      const int tap = rem - ic * 9;
      lds_w[((tap * (KC / 2) + (ic >> 1)) * 64 + oc) * 2 + (ic & 1)] =
          w[wbase + (size_t)oc * Cin * 9 + (size_t)(kc0 + ic) * 9 + tap];
    }
    // ---- stage input: KC ch x 3 rows x 130 cols, zero-padded halo ----------
    for (int e = tid; e < KC * 390; e += 256) {
      const int ic     = e / 390;
      const int rem    = e - ic * 390;
      const int row    = rem / 130;
      const int colIdx = rem - row * 130;      // 0..129 -> col = colIdx-1
      const int col    = colIdx - 1;
      const int gr     = h0 + row - 1;
      float v = 0.f;
      if (gr >= 0 && gr < Hd && col >= 0 && col < Wd)
        v = xn[(size_t)(kc0 + ic) * HWSZ + gr * Wd + col];
      lds_x[(((ic >> 1) * 3 + row) * 130 + colIdx) * 2 + (ic & 1)] = v;
    }
    __syncthreads();

    // ---- compute: 9 taps x (KC/4) K-steps, 4 WMMAs each --------------------
    for (int tap = 0; tap < 9; ++tap) {
      const int r = tap / 3, s = tap % 3;      // 0..2
#pragma unroll
      for (int kcs = 0; kcs < KC; kcs += 4) {
        const int aslot = ((kcs >> 1) + half) * 3 + r;
        const v2f a = *(const v2f*)&lds_x[(aslot * 130 + colbase + colc + s) * 2];
        const int wslot = (tap * (KC / 2) + (kcs >> 1) + half) * 64;
        const v2f b0 = *(const v2f*)&lds_w[(wslot + colc) * 2];
        acc0 = __builtin_amdgcn_wmma_f32_16x16x4_f32(false, a, false, b0, (short)0, acc0, false, false);
        const v2f b1 = *(const v2f*)&lds_w[(wslot + 16 + colc) * 2];
        acc1 = __builtin_amdgcn_wmma_f32_16x16x4_f32(false, a, false, b1, (short)0, acc1, false, false);
        const v2f b2 = *(const v2f*)&lds_w[(wslot + 32 + colc) * 2];
        acc2 = __builtin_amdgcn_wmma_f32_16x16x4_f32(false, a, false, b2, (short)0, acc2, false, false);
        const v2f b3 = *(const v2f*)&lds_w[(wslot + 48 + colc) * 2];
        acc3 = __builtin_amdgcn_wmma_f32_16x16x4_f32(false, a, false, b3, (short)0, acc3, false, false);
      }
    }
  }

  // ---- store D + bias as aligned float4 pairs ------------------------------
  const int rowbase = half * 8;
  const v8f accs[4] = {acc0, acc1, acc2, acc3};
#pragma unroll
  for (int j = 0; j < 4; ++j) {
    const int ch   = ngrp * 64 + j * 16 + colc;
    const float bv = bias[ch];
    float* op = out + ((size_t)n * O + ch) * HWSZ + h0 * Wd + colbase + rowbase;
    const v8f a = accs[j];
    float4 q0 = make_float4(a[0] + bv, a[1] + bv, a[2] + bv, a[3] + bv);
    float4 q1 = make_float4(a[4] + bv, a[5] + bv, a[6] + bv, a[7] + bv);
    *(float4*)(op)     = q0;
    *(float4*)(op + 4) = q1;
  }
}

// ---------------------------------------------------------------------------
// per-channel sum / sum-of-squares over [Nimg, Ch, HWSZ]; one block per channel
// ---------------------------------------------------------------------------
__global__ void chan_stats(const float* __restrict__ x, float* __restrict__ stats,
                           int Nimg, int Ch) {
  const int c = blockIdx.x;
  float s = 0.f, s2 = 0.f;
  for (int n = 0; n < Nimg; ++n) {
    const float* p = x + ((size_t)n * Ch + c) * HWSZ;
    for (int i = threadIdx.x; i < HWSZ; i += blockDim.x) {
      const float v = p[i];
      s += v; s2 += v * v;
    }
  }
  __shared__ float sh[512];
  sh[threadIdx.x] = s;
  sh[threadIdx.x + 256] = s2;
  __syncthreads();
  for (int o = 128; o > 0; o >>= 1) {
    if ((int)threadIdx.x < o) {
      sh[threadIdx.x] += sh[threadIdx.x + o];
      sh[256 + threadIdx.x] += sh[256 + threadIdx.x + o];
    }
    __syncthreads();
  }
  if (threadIdx.x == 0) { stats[2 * c] = sh[0]; stats[2 * c + 1] = sh[256]; }
}

// ---------------------------------------------------------------------------
// fused BN(train)+ReLU + ragged segment max   y:[28,256,HWSZ] -> pooled:[8,256,HWSZ]
// ---------------------------------------------------------------------------
__global__ void bn_relu_segmax(const float* __restrict__ y, const float* __restrict__ stats,
                               const float* __restrict__ gamma, const float* __restrict__ beta,
                               float invcnt, float* __restrict__ pooled) {
  const int idx = blockIdx.x * blockDim.x + threadIdx.x;
  if (idx >= 8 * 256 * HWSZ) return;
  const int p = idx & (HWSZ - 1);
  const int c = (idx >> 13) & 255;
  const int g = idx >> 21;
  const float mean = stats[2 * c] * invcnt;
  const float var  = stats[2 * c + 1] * invcnt - mean * mean;
  const float sc = gamma[c] * rsqrtf(var + EPS);
  const float sh = beta[c] - mean * sc;
  float mx = -3.402823e38f;
  const int e = c_seg_off[g + 1];
  for (int n = c_seg_off[g]; n < e; ++n) {
    float v = y[((size_t)n * 256 + c) * HWSZ + p] * sc + sh;
    mx = fmaxf(mx, fmaxf(v, 0.f));
  }
  pooled[idx] = mx;
}

// ---------------------------------------------------------------------------
// hist[bs] = max_{t=1..3} pooled[t,bs];  sub[bs] = pooled[t=bs,b=0] - hist[bs]
// (min_t(base - pooled[t]) == base - max_t pooled[t])
// ---------------------------------------------------------------------------
__global__ void hist_sub(const float* __restrict__ pooled,
                         float* __restrict__ hist, float* __restrict__ sub) {
  const int idx = blockIdx.x * blockDim.x + threadIdx.x;
  if (idx >= 2 * 256 * HWSZ) return;
  const int bs = idx >> 21;                 // 256*HWSZ = 2^21
  const int rem = idx & ((1 << 21) - 1);
  const size_t CS = (size_t)256 * HWSZ;
  const float h1 = pooled[(size_t)(2 + bs) * CS + rem];
  const float h2 = pooled[(size_t)(4 + bs) * CS + rem];
  const float h3 = pooled[(size_t)(6 + bs) * CS + rem];
  const float hmax = fmaxf(h1, fmaxf(h2, h3));
  const float base = pooled[(size_t)(bs * 2) * CS + rem];
  hist[idx] = hmax;
  sub[idx]  = base - hmax;
}

// ---------------------------------------------------------------------------
// BN(train) finalize + activation (mode 0 = ReLU, 1 = sigmoid)
// ---------------------------------------------------------------------------
__global__ void bn_act(const float* __restrict__ raw, const float* __restrict__ stats,
                       const float* __restrict__ gamma, const float* __restrict__ beta,
                       float invcnt, int Ch, int total, float* __restrict__ out, int mode) {
  const int idx = blockIdx.x * blockDim.x + threadIdx.x;
  if (idx >= total) return;
  const int c = (idx >> 13) % Ch;
  const float mean = stats[2 * c] * invcnt;
  const float var  = stats[2 * c + 1] * invcnt - mean * mean;
  const float sc = gamma[c] * rsqrtf(var + EPS);
  const float sh = beta[c] - mean * sc;
  const float v = raw[idx] * sc + sh;
  out[idx] = mode ? (1.f / (1.f + __expf(-v))) : fmaxf(v, 0.f);
}

// ---------------------------------------------------------------------------
// small scalar conv3x3 for the 64 -> 1 mask head
// ---------------------------------------------------------------------------
__global__ void conv3x3_small(const float* __restrict__ x, const float* __restrict__ w,
                              const float* __restrict__ bias, float* __restrict__ out,
                              int Nimg, int Cin) {
  const int idx = blockIdx.x * blockDim.x + threadIdx.x;
  if (idx >= Nimg * HWSZ) return;
  const int n = idx >> 13;
  const int p = idx & (HWSZ - 1);
  const int h = p >> 7, ww = p & 127;
  float acc = bias[0];
  for (int ci = 0; ci < Cin; ++ci) {
    const float* xc = x + ((size_t)n * Cin + ci) * HWSZ;
    const float* wc = w + ci * 9;
#pragma unroll
    for (int t = 0; t < 9; ++t) {
      const int r = t / 3 - 1, s = t % 3 - 1;
      const int hh = h + r, w2 = ww + s;
      if (hh >= 0 && hh < Hd && w2 >= 0 && w2 < Wd)
        acc += xc[hh * Wd + w2] * wc[t];
    }
  }
  out[idx] = acc;
}

// masked[bs,c,p] = hist_out[bs,c,p] * mask[bs,p]
__global__ void mask_mul(const float* __restrict__ hist_out, const float* __restrict__ mask,
                         float* __restrict__ out) {
  const int idx = blockIdx.x * blockDim.x + threadIdx.x;
  if (idx >= 2 * 256 * HWSZ) return;
  const int bs = idx >> 21;
  const int p = idx & (HWSZ - 1);
  out[idx] = hist_out[idx] * mask[bs * HWSZ + p];
}

extern "C" void kernel_launch(void* const* d_in, const int* in_sizes, int n_in,
                              void* d_out, int out_size, void* d_ws, size_t ws_size,
                              hipStream_t stream) {
  const float* feat   = (const float*)d_in[0];
  const float* w_cav  = (const float*)d_in[1];
  const float* b_cav  = (const float*)d_in[2];
  const float* g_cav  = (const float*)d_in[3];
  const float* be_cav = (const float*)d_in[4];
  const float* w_hist = (const float*)d_in[5];
  const float* b_hist = (const float*)d_in[6];
  const float* g_hist = (const float*)d_in[7];
  const float* be_hist= (const float*)d_in[8];
  const float* w_m1   = (const float*)d_in[9];
  const float* b_m1   = (const float*)d_in[10];
  const float* g_m1   = (const float*)d_in[11];
  const float* be_m1  = (const float*)d_in[12];
  const float* w_m2   = (const float*)d_in[13];
  const float* b_m2   = (const float*)d_in[14];
  const float* g_m2   = (const float*)d_in[15];
  const float* be_m2  = (const float*)d_in[16];
  const float* w_out  = (const float*)d_in[17];
  const float* b_out  = (const float*)d_in[18];
  const float* g_out  = (const float*)d_in[19];
  const float* be_out = (const float*)d_in[20];

  float* ws = (float*)d_ws;
  // high region (live across whole pipeline start)
  float* y1     = ws;                       // 28*256*8192 = 58,720,256
  float* pooled = ws + 58720256;            // 8*256*8192  = 16,777,216
  float* hist   = pooled + 16777216;        // 2*256*8192  =  4,194,304
  float* sub    = hist + 4194304;           //              4,194,304
  float* stats1 = sub + 4194304;            // 512
  // reused region (overlays y1 once pooled/hist/sub are computed)
  float* histc    = ws;                     // 4,194,304
  float* hist_out = ws + 4194304;           // 4,194,304
  float* mraw     = ws + 8388608;           // 1,048,576
  float* mact     = ws + 9437184;           // 1,048,576
  float* maskraw  = ws + 10485760;          // 16,384
  float* masked   = ws + 10502144;          // 4,194,304
  float* outraw   = ws + 14696448;          // 4,194,304
  float* stats_h  = ws + 18890752;          // 512
  float* stats_m1 = stats_h + 512;          // 128
  float* stats_m2 = stats_m1 + 128;         // 2
  float* stats_o  = stats_m2 + 16;          // 512

  float* mask_out  = (float*)d_out;         // [2, 8192]
  float* fused_out = (float*)d_out + 16384; // [2, 256, 8192]

  // 1) cav_fusion conv (dominant: ~270 GFLOP fp32 on the matrix engine)
  conv3x3_wmma<<<28 * 4 * Hd, 256, 0, stream>>>(feat, w_cav, b_cav, y1, 28, 256, 256);
  chan_stats<<<256, 256, 0, stream>>>(y1, stats1, 28, 256);
  { const int total = 8 * 256 * HWSZ;
    bn_relu_segmax<<<total / 256, 256, 0, stream>>>(y1, stats1, g_cav, be_cav,
                                                    1.f / (28.f * 8192.f), pooled); }
  { const int total = 2 * 256 * HWSZ;
    hist_sub<<<total / 256, 256, 0, stream>>>(pooled, hist, sub); }

  // 2) historical_fusion
  conv3x3_wmma<<<2 * 4 * Hd, 256, 0, stream>>>(hist, w_hist, b_hist, histc, 2, 256, 256);
  chan_stats<<<256, 256, 0, stream>>>(histc, stats_h, 2, 256);
  { const int total = 2 * 256 * HWSZ;
    bn_act<<<total / 256, 256, 0, stream>>>(histc, stats_h, g_hist, be_hist,
                                            1.f / 16384.f, 256, total, hist_out, 0); }

  // 3) mask_model stage 1 (256 -> 64)
  conv3x3_wmma<<<2 * 1 * Hd, 256, 0, stream>>>(sub, w_m1, b_m1, mraw, 2, 256, 64);
  chan_stats<<<64, 256, 0, stream>>>(mraw, stats_m1, 2, 64);
  { const int total = 2 * 64 * HWSZ;
    bn_act<<<total / 256, 256, 0, stream>>>(mraw, stats_m1, g_m1, be_m1,
                                            1.f / 16384.f, 64, total, mact, 0); }

  // 4) mask_model stage 2 (64 -> 1) + sigmoid -> mask output
  { const int total = 2 * HWSZ;
    conv3x3_small<<<total / 256, 256, 0, stream>>>(mact, w_m2, b_m2, maskraw, 2, 64); }
  chan_stats<<<1, 256, 0, stream>>>(maskraw, stats_m2, 2, 1);
  { const int total = 2 * HWSZ;
    bn_act<<<total / 256, 256, 0, stream>>>(maskraw, stats_m2, g_m2, be_m2,
                                            1.f / 16384.f, 1, total, mask_out, 1); }

  // 5) output_layer on hist_out * mask
  { const int total = 2 * 256 * HWSZ;
    mask_mul<<<total / 256, 256, 0, stream>>>(hist_out, mask_out, masked); }
  conv3x3_wmma<<<2 * 4 * Hd, 256, 0, stream>>>(masked, w_out, b_out, outraw, 2, 256, 256);
  chan_stats<<<256, 256, 0, stream>>>(outraw, stats_o, 2, 256);
  { const int total = 2 * 256 * HWSZ;
    bn_act<<<total / 256, 256, 0, stream>>>(outraw, stats_o, g_out, be_out,
                                            1.f / 16384.f, 256, total, fused_out, 0); }
}